// LevelWisePooling_14388140442145
// MI455X (gfx1250) — compile-verified
//
#include <hip/hip_runtime.h>

// LevelWisePooling: segment mean+max over 64 depth levels, D=128, N=2,000,000.
// HBM-bound streaming reduce (~1 GB read -> ~44 us floor @ 23.3 TB/s).
// Strategy: block-private LDS bins (ds_add_f32 / ds_max_u32, bank-conflict-free
// via component-order swizzle), non-temporal b128 streaming loads (unguarded
// unroll-8 main loop for ~8 MB in-flight), register counts, global-atomic
// merge, tiny finalize kernel. WMMA is inapplicable (reduction, not matmul).

#define NLEV   64
#define DIM    128
#define BLOCK  256   // 8 waves (wave32)
#define NBLK   256   // 2048 waves device-wide
#define UNROLL 8     // 8 b128 loads in flight per wave in the main loop

typedef float v4f __attribute__((ext_vector_type(4)));

// Order-preserving float -> u32 key (native integer max atomics, no CAS loop).
__device__ __forceinline__ unsigned encf(float x) {
    unsigned u = __float_as_uint(x);
    return (u & 0x80000000u) ? ~u : (u | 0x80000000u);
}
__device__ __forceinline__ float decf(unsigned k) {
    return (k & 0x80000000u) ? __uint_as_float(k & 0x7fffffffu)
                             : __uint_as_float(~k);
}

__global__ void __launch_bounds__(BLOCK)
init_bins(float* __restrict__ gsum, unsigned* __restrict__ gmax,
          float* __restrict__ gcnt) {
    int i = blockIdx.x * BLOCK + threadIdx.x;
    if (i < NLEV * DIM) { gsum[i] = 0.0f; gmax[i] = 0u; }  // key 0 <= every real key
    if (i < NLEV)       { gcnt[i] = 0.0f; }
}

__global__ void __launch_bounds__(BLOCK)
pool_kernel(const float* __restrict__ emb,
            const int*   __restrict__ dep2,   // int64 depths viewed as int pairs (LE)
            float* __restrict__ gsum, unsigned* __restrict__ gmax,
            float* __restrict__ gcnt, int nrows) {
    __shared__ float    lsum[NLEV * DIM];   // 32 KB
    __shared__ unsigned lmax[NLEV * DIM];   // 32 KB -> total exactly 64 KB

    const int tid = threadIdx.x;
    for (int i = tid; i < NLEV * DIM; i += BLOCK) { lsum[i] = 0.0f; lmax[i] = 0u; }
    __syncthreads();

    const int lane = tid & 31;
    const int wid  = tid >> 5;
    const long long stride = (long long)NBLK * (BLOCK / 32);            // 2048
    long long r0 = (long long)blockIdx.x * (BLOCK / 32) + wid;          // wave id
    // Bank-conflict-free swizzle: lanes 16..31 issue components in order
    // (2,3,0,1) so each ds op touches 32 distinct banks instead of 16 x 2-way.
    const int sw = (lane & 16) ? 2 : 0;

    float c0 = 0.0f, c1 = 0.0f;  // lane counts levels `lane` and `lane+32`

    // ---- Unguarded main loop: all UNROLL rows valid -> loads issue as one
    // clause, no per-row exec-mask juggling.
    const long long mainEnd = (long long)nrows - (UNROLL - 1) * stride;
    for (; r0 < mainEnd; r0 += stride * UNROLL) {
        v4f v[UNROLL];
        int dep[UNROLL];
#pragma unroll
        for (int u = 0; u < UNROLL; ++u) {
            const long long r = r0 + (long long)u * stride;
            dep[u] = dep2[2 * r];  // low dword of int64, value in [0,64)
            v[u] = __builtin_nontemporal_load(
                       (const v4f*)(emb + r * (long long)DIM) + lane);
        }
#pragma unroll
        for (int u = 0; u < UNROLL; ++u) {
            const int base = dep[u] * DIM + lane * 4;
#pragma unroll
            for (int j = 0; j < 4; ++j) {
                const int c = (j + sw) & 3;
                atomicAdd(&lsum[base + c], v[u][c]);        // ds_add_f32
                atomicMax(&lmax[base + c], encf(v[u][c]));  // ds_max_u32
            }
            c0 += (dep[u] == lane)      ? 1.0f : 0.0f;
            c1 += (dep[u] == lane + 32) ? 1.0f : 0.0f;
        }
    }

    // ---- Per-row tail (at most UNROLL-1 rows per wave).
    for (; r0 < nrows; r0 += stride) {
        const int depth = dep2[2 * r0];
        const v4f v = __builtin_nontemporal_load(
                          (const v4f*)(emb + r0 * (long long)DIM) + lane);
        const int base = depth * DIM + lane * 4;
#pragma unroll
        for (int j = 0; j < 4; ++j) {
            const int c = (j + sw) & 3;
            atomicAdd(&lsum[base + c], v[c]);
            atomicMax(&lmax[base + c], encf(v[c]));
        }
        c0 += (depth == lane)      ? 1.0f : 0.0f;
        c1 += (depth == lane + 32) ? 1.0f : 0.0f;
    }

    // Per-wave counts straight to global (32 distinct addresses per instr).
    atomicAdd(&gcnt[lane],      c0);
    atomicAdd(&gcnt[lane + 32], c1);

    __syncthreads();
    for (int i = tid; i < NLEV * DIM; i += BLOCK) {
        atomicAdd(&gsum[i], lsum[i]);   // global_atomic_add_f32
        atomicMax(&gmax[i], lmax[i]);   // global_atomic_max_u32
    }
}

__global__ void __launch_bounds__(BLOCK)
finalize_kernel(const float* __restrict__ gsum, const unsigned* __restrict__ gmax,
                const float* __restrict__ gcnt, float* __restrict__ out) {
    int i = blockIdx.x * BLOCK + threadIdx.x;
    if (i >= NLEV * DIM) return;
    const int l = i / DIM;
    const int d = i - l * DIM;
    const float cnt  = gcnt[l];
    const bool  live = (cnt > 0.0f);
    out[l * (2 * DIM) + d]       = live ? gsum[i] / cnt : 0.0f;  // mean
    out[l * (2 * DIM) + DIM + d] = live ? decf(gmax[i]) : 0.0f;  // max (empty -> 0)
}

extern "C" void kernel_launch(void* const* d_in, const int* in_sizes, int n_in,
                              void* d_out, int out_size, void* d_ws, size_t ws_size,
                              hipStream_t stream) {
    const float* emb  = (const float*)d_in[0];
    const int*   dep2 = (const int*)d_in[1];   // int64 depths, read low dwords
    const int nrows = in_sizes[1];             // 2,000,000

    // Workspace layout: sums[8192] f32 | maxkeys[8192] u32 | counts[64] f32
    float*    gsum = (float*)d_ws;
    unsigned* gmax = (unsigned*)(gsum + NLEV * DIM);
    float*    gcnt = (float*)(gmax + NLEV * DIM);
    float*    out  = (float*)d_out;

    const int binBlocks = (NLEV * DIM + BLOCK - 1) / BLOCK;  // 32
    init_bins<<<binBlocks, BLOCK, 0, stream>>>(gsum, gmax, gcnt);
    pool_kernel<<<NBLK, BLOCK, 0, stream>>>(emb, dep2, gsum, gmax, gcnt, nrows);
    finalize_kernel<<<binBlocks, BLOCK, 0, stream>>>(gsum, gmax, gcnt, out);
}